// SphericalHarmonicsEncoding_74809740362342
// MI455X (gfx1250) — compile-verified
//
#include <hip/hip_runtime.h>
#include <hip/hip_bf16.h>

typedef __attribute__((ext_vector_type(2))) float v2f;
typedef __attribute__((ext_vector_type(8))) float v8f;

// ---------------------------------------------------------------------------
// out[n] = sum_k mono[k] * B[k][n], monomial order k:
// 1, x,y,z, x2,y2,z2,xy,xz,yz, x3,y3,z3, x2y,x2z,xy2,y2z,xz2,yz2,xyz,
// x4,y4,z4, x3y,x3z,xy3,y3z,xz3,yz3, x2y2,x2z2,y2z2, x2yz,xy2z,xyz2, pad.
// Signs follow the reference convention (Condon-Shortley phase kept, no
// extra (-1)^m), so Y(l,+/-1) and Y(l,+/-3) are negated vs graphics tables.
// ---------------------------------------------------------------------------
namespace {
struct BMatHost { float v[36][32]; };
constexpr BMatHost makeB() {
    BMatHost b{};  // zero-init
    // l = 0
    b.v[0][0]   =  0.28209479177387814f;
    // l = 1  (m=-1,0,1)
    b.v[2][1]   = -0.48860251190291992f;                      // -c*y
    b.v[3][2]   =  0.48860251190291992f;                      //  c*z
    b.v[1][3]   = -0.48860251190291992f;                      // -c*x
    // l = 2  (m=-2..2)
    b.v[7][4]   =  1.09254843059207920f;                      //  c*xy
    b.v[9][5]   = -1.09254843059207920f;                      // -c*yz
    b.v[6][6]   =  0.94617469575756008f;                      //  c*(3z^2-1)
    b.v[0][6]   = -0.31539156525252005f;
    b.v[8][7]   = -1.09254843059207920f;                      // -c*xz
    b.v[4][8]   =  0.54627421529603959f;                      //  c*(x^2-y^2)
    b.v[5][8]   = -0.54627421529603959f;
    // l = 3  (m=-3..3)
    b.v[13][9]  = -1.77013076977993040f;                      // -c*y(3x^2-y^2)
    b.v[11][9]  =  0.59004358992664352f;
    b.v[19][10] =  2.89061144264055380f;                      //  c*xyz
    b.v[18][11] = -2.28522899732232880f;                      // -c*y(5z^2-1)
    b.v[2][11]  =  0.45704579946446577f;
    b.v[12][12] =  1.86588166295057690f;                      //  c*(5z^3-3z)
    b.v[3][12]  = -1.11952899777034620f;
    b.v[17][13] = -2.28522899732232880f;                      // -c*x(5z^2-1)
    b.v[1][13]  =  0.45704579946446577f;
    b.v[14][14] =  1.44530572132027700f;                      //  c*z(x^2-y^2)
    b.v[16][14] = -1.44530572132027700f;
    b.v[10][15] = -0.59004358992664352f;                      // -c*x(x^2-3y^2)
    b.v[15][15] =  1.77013076977993040f;
    // l = 4  (m=-4..4)
    b.v[23][16] =  2.50334294179670460f;                      //  c*xy(x^2-y^2)
    b.v[25][16] = -2.50334294179670460f;
    b.v[32][17] = -5.31039230933979130f;                      // -c*yz(3x^2-y^2)
    b.v[26][17] =  1.77013076977993040f;
    b.v[34][18] =  6.62322287030292070f;                      //  c*xy(7z^2-1)
    b.v[7][18]  = -0.94617469575756008f;
    b.v[28][19] = -4.68332580490102440f;                      // -c*yz(7z^2-3)
    b.v[9][19]  =  2.00713963067186760f;
    b.v[22][20] =  3.70249414203215100f;                      //  c*(35z^4-30z^2+3)
    b.v[6][20]  = -3.17356640745612930f;
    b.v[0][20]  =  0.31735664074561293f;
    b.v[27][21] = -4.68332580490102440f;                      // -c*xz(7z^2-3)
    b.v[8][21]  =  2.00713963067186760f;
    b.v[30][22] =  3.31161143515146030f;                      //  c*(x^2-y^2)(7z^2-1)
    b.v[4][22]  = -0.47308734787878004f;
    b.v[31][22] = -3.31161143515146030f;
    b.v[5][22]  =  0.47308734787878004f;
    b.v[24][23] = -1.77013076977993040f;                      // -c*xz(x^2-3y^2)
    b.v[33][23] =  5.31039230933979130f;
    b.v[20][24] =  0.62583573544917614f;                      //  c*(x^4-6x^2y^2+y^4)
    b.v[29][24] = -3.75501441269505630f;
    b.v[21][24] =  0.62583573544917614f;
    return b;
}

// Pre-swizzled into WMMA B-fragment order so each lane's fragment is one
// contiguous 8-byte pair:  v[g][half][n] = { B[4g+2*half][n], B[4g+2*half+1][n] }
// -> one global_load_b64 per (nt,g) from a single base pointer.
struct BSwz { float v[9][2][32][2]; };
constexpr BSwz makeBSwz() {
    BSwz s{};
    BMatHost b = makeB();
    for (int g = 0; g < 9; ++g)
        for (int h = 0; h < 2; ++h)
            for (int n = 0; n < 32; ++n) {
                const int k = 4 * g + 2 * h;
                s.v[g][h][n][0] = b.v[k][n];
                s.v[g][h][n][1] = b.v[k + 1][n];
            }
    return s;
}
} // namespace

__device__ __constant__ __align__(16) BSwz g_B = makeBSwz();

#define LDS_STRIDE 38   // even, gcd(38,64)=2: conflict-free b64 LDS, 8B-aligned rows

__global__ __launch_bounds__(256)
void SphericalHarmonicsEncoding_74809740362342_kernel(
        const float* __restrict__ dirs, float* __restrict__ out, int npts) {
    __shared__ __align__(16) float sm[256 * LDS_STRIDE];

    const int tid       = threadIdx.x;
    const int lane      = tid & 31;
    const int wave      = tid >> 5;
    const int half      = lane >> 4;   // 0 or 1
    const int l16       = lane & 15;
    const int blockBase = blockIdx.x * 256;

    // -------- B fragments first: 18 clause-able b64 loads, single base ------
    // byte offset: g*512 + nt*128 + (half*32 + l16)*8
    const char* Bbase = (const char*)&g_B + (size_t)((half * 32 + l16) * 8);
    v2f bFrag[2][9];
#pragma unroll
    for (int nt = 0; nt < 2; ++nt) {
#pragma unroll
        for (int g = 0; g < 9; ++g) {
            bFrag[nt][g] = *reinterpret_cast<const v2f*>(Bbase + g * 512 + nt * 128);
        }
    }

    // -------- per-lane: load direction, normalize, build 35 monomials -------
    int p = blockBase + tid;
    if (p > npts - 1) p = npts - 1;      // clamp, never diverge EXEC
    float x = dirs[3 * p + 0];
    float y = dirs[3 * p + 1];
    float z = dirs[3 * p + 2];
    float nrm = sqrtf(x * x + y * y + z * z);
    nrm = fmaxf(nrm, 1e-12f);
    float inv = 1.0f / nrm;
    x *= inv; y *= inv; z *= inv;

    float m[36];
    const float x2 = x * x, y2 = y * y, z2 = z * z;
    const float xy = x * y, xz = x * z, yz = y * z;
    m[0]  = 1.0f; m[1]  = x;  m[2]  = y;  m[3]  = z;
    m[4]  = x2;   m[5]  = y2; m[6]  = z2; m[7]  = xy; m[8] = xz; m[9] = yz;
    m[10] = x2 * x;  m[11] = y2 * y;  m[12] = z2 * z;
    m[13] = x2 * y;  m[14] = x2 * z;  m[15] = x * y2; m[16] = y2 * z;
    m[17] = x * z2;  m[18] = y * z2;  m[19] = xy * z;
    m[20] = x2 * x2; m[21] = y2 * y2; m[22] = z2 * z2;
    m[23] = m[10] * y; m[24] = m[10] * z; m[25] = x * m[11]; m[26] = m[11] * z;
    m[27] = x * m[12]; m[28] = y * m[12];
    m[29] = x2 * y2;   m[30] = x2 * z2;  m[31] = y2 * z2;
    m[32] = x2 * yz;   m[33] = y2 * xz;  m[34] = z2 * xy;
    m[35] = 0.0f;

    // -------- stage monomials in LDS (row per lane, 8B-aligned pairs) -------
    float* row = &sm[tid * LDS_STRIDE];
#pragma unroll
    for (int k = 0; k < 36; k += 2) {
        *reinterpret_cast<v2f*>(&row[k]) = v2f{m[k], m[k + 1]};
    }
    __syncthreads();

    // -------- A fragments: 16x4 f32 layout (V0: K=0|K=2, V1: K=1|K=3) ------
    v2f aFrag[2][9];
#pragma unroll
    for (int mt = 0; mt < 2; ++mt) {
        const int srcRow = wave * 32 + mt * 16 + l16;
        const float* r = &sm[srcRow * LDS_STRIDE];
#pragma unroll
        for (int g = 0; g < 9; ++g) {
            aFrag[mt][g] = *reinterpret_cast<const v2f*>(&r[4 * g + 2 * half]);
        }
    }

    // -------- accumulate: 36 x V_WMMA_F32_16X16X4_F32 per 32 points --------
    v8f acc[2][2];
#pragma unroll
    for (int mt = 0; mt < 2; ++mt)
#pragma unroll
        for (int nt = 0; nt < 2; ++nt)
            acc[mt][nt] = v8f{0.f, 0.f, 0.f, 0.f, 0.f, 0.f, 0.f, 0.f};

#pragma unroll
    for (int g = 0; g < 9; ++g) {
#pragma unroll
        for (int mt = 0; mt < 2; ++mt) {
#pragma unroll
            for (int nt = 0; nt < 2; ++nt) {
                acc[mt][nt] = __builtin_amdgcn_wmma_f32_16x16x4_f32(
                    /*neg_a=*/false, aFrag[mt][g],
                    /*neg_b=*/false, bFrag[nt][g],
                    /*c_mod=*/(short)0, acc[mt][nt],
                    /*reuse_a=*/false, /*reuse_b=*/false);
            }
        }
    }

    // -------- store D: VGPR j -> row M=j (lanes 0-15) / M=j+8 (16-31) ------
#pragma unroll
    for (int mt = 0; mt < 2; ++mt) {
        const int gpBase = blockBase + wave * 32 + mt * 16 + 8 * half;
#pragma unroll
        for (int nt = 0; nt < 2; ++nt) {
            const int col = nt * 16 + l16;
            if (col < 25) {
#pragma unroll
                for (int j = 0; j < 8; ++j) {
                    int gp = gpBase + j;
                    if (gp > npts - 1) gp = npts - 1;
                    out[(size_t)gp * 25 + col] = acc[mt][nt][j];
                }
            }
        }
    }
}

extern "C" void kernel_launch(void* const* d_in, const int* in_sizes, int n_in,
                              void* d_out, int out_size, void* d_ws, size_t ws_size,
                              hipStream_t stream) {
    const float* dirs = (const float*)d_in[0];
    float* out        = (float*)d_out;
    const int npts    = in_sizes[0] / 3;            // directions are [N,3]
    const int nblocks = (npts + 255) / 256;         // 256 thr = 8 waves, 32 pts/wave
    SphericalHarmonicsEncoding_74809740362342_kernel<<<nblocks, 256, 0, stream>>>(
        dirs, out, npts);
}